// S2ConvNet_SO3_6_19567871001313
// MI455X (gfx1250) — compile-verified
//
#include <hip/hip_runtime.h>

typedef float v2f __attribute__((ext_vector_type(2)));
typedef float v8f __attribute__((ext_vector_type(8)));

#define PI_D   3.14159265358979323846
#define TWO_PI 6.28318530717958647692

__device__ __forceinline__ int imin(int a, int b) { return a < b ? a : b; }
__device__ __forceinline__ int imax(int a, int b) { return a > b ? a : b; }

// ---------------- Wigner-d via log-Gamma Jacobi sum (double precision) ----------------
__device__ double wig_d(const double* lft, int l, int mm, int nn, double beta) {
  if (mm < -l || mm > l || nn < -l || nn > l) return 0.0;
  double ch = cos(0.5 * beta), sh = sin(0.5 * beta);
  int smin = imax(0, nn - mm);
  int smax = imin(l + nn, l - mm);
  double pref = 0.5 * (lft[l + mm] + lft[l - mm] + lft[l + nn] + lft[l - nn]);
  double acc = 0.0;
  for (int s = smin; s <= smax; ++s) {
    double lg = pref - lft[l + nn - s] - lft[s] - lft[mm - nn + s] - lft[l - mm - s];
    double t = exp(lg) * pow(ch, (double)(2 * l + nn - mm - 2 * s)) * pow(sh, (double)(mm - nn + 2 * s));
    acc += ((mm - nn + s) & 1) ? -t : t;
  }
  return acc;
}
__device__ __forceinline__ double qbeta(int j, int b) { return PI_D * (2.0 * j + 1.0) / (4.0 * b); }
__device__ double dhw(int j, int b) {
  double be = qbeta(j, b), s = 0.0;
  for (int k = 0; k < b; ++k) s += sin((2.0 * k + 1.0) * be) / (2.0 * k + 1.0);
  return (2.0 / b) * sin(be) * s;
}

__global__ void k_lftab(double* lft) {
  int i = threadIdx.x;
  if (i < 160) lft[i] = lgamma((double)i + 1.0);
}

// ---------------- constant builders ----------------
// W_S2 layout [m(59)][l(30)][j(60)]
__global__ void k_ws2(const double* lft, float* out) {
  long idx = (long)blockIdx.x * 256 + threadIdx.x;
  if (idx >= 59L * 30 * 60) return;
  int j = idx % 60; int l = (idx / 60) % 30; int m = (int)(idx / 1800);
  out[idx] = (float)(wig_d(lft, l, m - 29, 0, qbeta(j, 30)) * dhw(j, 30));
}
// DL_S2 layout [m][n][j(60)][l(30)]
__global__ void k_dls2(const double* lft, float* out) {
  long idx = (long)blockIdx.x * 256 + threadIdx.x;
  if (idx >= 59L * 59 * 60 * 30) return;
  int l = idx % 30; int j = (idx / 30) % 60; int n = (idx / 1800) % 59; int m = (int)(idx / (1800L * 59));
  out[idx] = (float)(wig_d(lft, l, m - 29, n - 29, qbeta(j, 30)) * (2.0 * l + 1.0));
}
// khat for S2: [o(2)][l(30)][n(59)] interleaved complex
__global__ void k_khs2(const float* __restrict__ ks2, const double* lft, float* __restrict__ out) {
  long idx = (long)blockIdx.x * 256 + threadIdx.x;
  if (idx >= 2L * 30 * 59) return;
  int n = idx % 59; int l = (idx / 59) % 30; int o = (int)(idx / (59 * 30));
  int np = n - 29;
  double dg = wig_d(lft, l, np, 0, PI_D / 16.0);
  float re = 0.f, im = 0.f;
  for (int p = 0; p < 6; ++p) {
    double ang = -(TWO_PI / 6.0) * (double)(np * p);
    float wv = ks2[o * 6 + p];
    re += wv * (float)cos(ang);
    im += wv * (float)sin(ang);
  }
  out[idx * 2] = (float)(re * dg);
  out[idx * 2 + 1] = (float)(im * dg);
}
// Wd layout [m][k][l(bo)][j(2bi)]
__global__ void k_wd_so3(const double* lft, float* out, int bi, int bo) {
  int Mc = 2 * bo - 1;
  long total = (long)Mc * Mc * bo * 2 * bi;
  long idx = (long)blockIdx.x * 256 + threadIdx.x;
  if (idx >= total) return;
  int j = idx % (2 * bi); long r = idx / (2 * bi);
  int l = r % bo; r /= bo;
  int k = r % Mc; r /= Mc;
  int m = (int)r;
  out[idx] = (float)(wig_d(lft, l, m - (bo - 1), k - (bo - 1), qbeta(j, bi)) * dhw(j, bi));
}
// DL layout [m][n][j(2bo)][l(bo)]
__global__ void k_dl_so3(const double* lft, float* out, int bo) {
  int Mc = 2 * bo - 1;
  long total = (long)Mc * Mc * 2 * bo * bo;
  long idx = (long)blockIdx.x * 256 + threadIdx.x;
  if (idx >= total) return;
  int l = idx % bo; long r = idx / bo;
  int j = r % (2 * bo); r /= (2 * bo);
  int n = r % Mc; r /= Mc;
  int m = (int)r;
  out[idx] = (float)(wig_d(lft, l, m - (bo - 1), n - (bo - 1), qbeta(j, bo)) * (2.0 * l + 1.0));
}
// khat for SO3 in split planes, B-matrix layout: [l][plane][ (i*Mc+k) ][ (o*Mc+n) ]
__global__ void k_khat_so3(const float* __restrict__ kin, const double* lft, float* __restrict__ out,
                           int bo, int ci, int co) {
  int Mc = 2 * bo - 1;
  long total = (long)bo * ci * Mc * co * Mc;
  long idx = (long)blockIdx.x * 256 + threadIdx.x;
  if (idx >= total) return;
  int n = idx % Mc; long r = idx / Mc;
  int o = r % co; r /= co;
  int k = r % Mc; r /= Mc;
  int i = r % ci; r /= ci;
  int l = (int)r;
  int np = n - (bo - 1), kp = k - (bo - 1);
  double dg = wig_d(lft, l, np, kp, PI_D / 16.0);
  float re = 0.f, im = 0.f;
  for (int p = 0; p < 6; ++p)
    for (int q = 0; q < 6; ++q) {
      float wv = kin[(i * co + o) * 36 + p * 6 + q];
      double ang = -(TWO_PI / 6.0) * (double)(np * p + kp * q);
      re += wv * (float)cos(ang);
      im += wv * (float)sin(ang);
    }
  re *= (float)dg; im *= (float)dg;
  long K = (long)ci * Mc, N = (long)co * Mc;
  long b0 = ((long)(l * 2 + 0) * K + i * Mc + k) * N + o * Mc + n;
  long b1 = ((long)(l * 2 + 1) * K + i * Mc + k) * N + o * Mc + n;
  out[b0] = re; out[b1] = im;
}

// ---------------- DFT kernels (direct, small non-pow2 sizes) ----------------
// x (32,1,60,60) -> xfc [m(59)][j(60)][b(32)] complex, scaled by 2pi/60
__global__ void k_dft_s2(const float* __restrict__ x, float* __restrict__ out) {
  long idx = (long)blockIdx.x * 256 + threadIdx.x;
  if (idx >= 59L * 60 * 32) return;
  int b = idx % 32; int j = (idx / 32) % 60; int m = (int)(idx / (60 * 32));
  int mp = m - 29;
  const float* xr = x + (long)b * 3600 + j * 60;
  float re = 0.f, im = 0.f;
  for (int a = 0; a < 60; ++a) {
    float ang = -(float)TWO_PI * mp * a / 60.0f;
    float s, c; sincosf(ang, &s, &c);
    re += xr[a] * c; im += xr[a] * s;
  }
  float sc = (float)(TWO_PI / 60.0);
  out[idx * 2] = re * sc; out[idx * 2 + 1] = im * sc;
}
// fhat[m][n][l][b][o] cplx = xhat[m][l][b] * khat[o][l][n]
__global__ void k_fhat_s2(const float* __restrict__ xh, const float* __restrict__ kh, float* __restrict__ fh) {
  long idx = (long)blockIdx.x * 256 + threadIdx.x;
  if (idx >= 59L * 59 * 30 * 32 * 2) return;
  int o = idx % 2; long r = idx / 2;
  int b = r % 32; r /= 32;
  int l = r % 30; r /= 30;
  int n = r % 59; r /= 59;
  int m = (int)r;
  float xr = xh[((long)m * 30 + l) * 64 + b * 2];
  float xi = xh[((long)m * 30 + l) * 64 + b * 2 + 1];
  const float* k2 = kh + ((long)(o * 30 + l) * 59 + n) * 2;
  fh[idx * 2] = xr * k2[0] - xi * k2[1];
  fh[idx * 2 + 1] = xr * k2[1] + xi * k2[0];
}
// h[b][c][j][a][g] (real) -> U[b][c][j][a][k(Mc)] cplx over gamma, scaled
__global__ void k_dft_gamma(const float* __restrict__ h, float* __restrict__ U,
                            int N, int C, int Mc, float scale) {
  long total = 32L * C * N * N * Mc;
  long idx = (long)blockIdx.x * 256 + threadIdx.x;
  if (idx >= total) return;
  int k = idx % Mc; long r = idx / Mc;
  int a = r % N; r /= N;
  int j = r % N; r /= N;
  int c = r % C; r /= C;
  int b = (int)r;
  int kp = k - (Mc - 1) / 2;
  const float* hp = h + ((((long)(b * C + c) * N + j) * N + a) * N);
  float re = 0.f, im = 0.f;
  for (int g = 0; g < N; ++g) {
    float ang = -(float)TWO_PI * kp * g / (float)N;
    float s, cc; sincosf(ang, &s, &cc);
    re += hp[g] * cc; im += hp[g] * s;
  }
  U[idx * 2] = re * scale; U[idx * 2 + 1] = im * scale;
}
// U -> xfc [m][k][j][b][c] cplx over alpha
__global__ void k_dft_alpha2(const float* __restrict__ U, float* __restrict__ out, int N, int C, int Mc) {
  long total = (long)Mc * Mc * N * 32 * C;
  long idx = (long)blockIdx.x * 256 + threadIdx.x;
  if (idx >= total) return;
  int c = idx % C; long r = idx / C;
  int b = r % 32; r /= 32;
  int j = r % N; r /= N;
  int k = r % Mc; r /= Mc;
  int m = (int)r;
  int mp = m - (Mc - 1) / 2;
  float re = 0.f, im = 0.f;
  for (int a = 0; a < N; ++a) {
    const float* u = U + ((((long)(b * C + c) * N + j) * N + a) * Mc + k) * 2;
    float ang = -(float)TWO_PI * mp * a / (float)N;
    float s, cc; sincosf(ang, &s, &cc);
    re += u[0] * cc - u[1] * s;
    im += u[0] * s + u[1] * cc;
  }
  out[idx * 2] = re; out[idx * 2 + 1] = im;
}
// inverse over n: G[m][n][j][bc] -> T[m][g][j][bc]
__global__ void k_idft_n(const float* __restrict__ G, float* __restrict__ T, int N, int BC) {
  int Mc = N - 1;
  long total = (long)Mc * N * N * BC;
  long idx = (long)blockIdx.x * 256 + threadIdx.x;
  if (idx >= total) return;
  int bc = idx % BC; long r = idx / BC;
  int j = r % N; r /= N;
  int g = r % N; r /= N;
  int m = (int)r;
  float re = 0.f, im = 0.f;
  for (int n = 0; n < Mc; ++n) {
    const float* Gp = G + (((long)(m * Mc + n) * N + j) * BC + bc) * 2;
    int np = n - (Mc - 1) / 2;
    float ang = (float)TWO_PI * np * g / (float)N;
    float s, cc; sincosf(ang, &s, &cc);
    re += Gp[0] * cc - Gp[1] * s;
    im += Gp[0] * s + Gp[1] * cc;
  }
  T[idx * 2] = re; T[idx * 2 + 1] = im;
}
// inverse over m + Re + ReLU: T[m][g][j][bc] -> h[b][c][j][a][g]
__global__ void k_idft_m_relu(const float* __restrict__ T, float* __restrict__ h, int N, int C) {
  int Mc = N - 1;
  long total = 32L * C * N * N * N;
  long idx = (long)blockIdx.x * 256 + threadIdx.x;
  if (idx >= total) return;
  int g = idx % N; long r = idx / N;
  int a = r % N; r /= N;
  int j = r % N; r /= N;
  int c = r % C; r /= C;
  int b = (int)r;
  int bc = b * C + c, BC = 32 * C;
  float acc = 0.f;
  for (int m = 0; m < Mc; ++m) {
    const float* t = T + (((long)(m * N + g) * N + j) * BC + bc) * 2;
    int mp = m - (Mc - 1) / 2;
    float ang = (float)TWO_PI * mp * a / (float)N;
    float s, cc; sincosf(ang, &s, &cc);
    acc += t[0] * cc - t[1] * s;
  }
  h[idx] = fmaxf(acc, 0.f);
}

// ---------------- WMMA GEMMs (fp32, V_WMMA_F32_16X16X4_F32) ----------------
__device__ __forceinline__ v8f wmma4(v2f a, v2f b, v8f c) {
  return __builtin_amdgcn_wmma_f32_16x16x4_f32(false, a, false, b, (short)0, c, false, false);
}
__device__ __forceinline__ v2f ldAf(const float* S, int tm, int lm, int ko) {
  v2f v; v.x = S[(tm * 16 + lm) * 33 + ko]; v.y = S[(tm * 16 + lm) * 33 + ko + 1]; return v;
}
__device__ __forceinline__ v2f ldBf(const float* S, int tn, int lm, int ko) {
  v2f v; v.x = S[ko * 65 + tn * 16 + lm]; v.y = S[(ko + 1) * 65 + tn * 16 + lm]; return v;
}

// batched real GEMM: C[z] = A[z](MxK) * B[z](KxN), row-major
__global__ __launch_bounds__(256) void k_gemm_rr(const float* __restrict__ A, const float* __restrict__ B,
                                                 float* __restrict__ C, int M, int N, int K,
                                                 long sA, long sB, long sC) {
  __shared__ float As[64 * 33];
  __shared__ float Bs[32 * 65];
  const float* Ab = A + (long)blockIdx.z * sA;
  const float* Bb = B + (long)blockIdx.z * sB;
  float* Cb = C + (long)blockIdx.z * sC;
  int m0 = blockIdx.y * 64, n0 = blockIdx.x * 64;
  int tid = threadIdx.x, lane = tid & 31, w = tid >> 5;
  int half = lane >> 4, lm = lane & 15;
  int t0 = w, t1 = w + 8;
  int t0m = t0 >> 2, t0n = t0 & 3, t1m = t1 >> 2, t1n = t1 & 3;
  v8f acc0 = {0, 0, 0, 0, 0, 0, 0, 0}, acc1 = {0, 0, 0, 0, 0, 0, 0, 0};
  for (int kc = 0; kc < K; kc += 32) {
    for (int i = tid; i < 64 * 32; i += 256) {
      int rr = i >> 5, cc = i & 31;
      int gm = m0 + rr, gk = kc + cc;
      As[rr * 33 + cc] = (gm < M && gk < K) ? Ab[(long)gm * K + gk] : 0.f;
    }
    for (int i = tid; i < 32 * 64; i += 256) {
      int rr = i >> 6, cc = i & 63;
      int gk = kc + rr, gn = n0 + cc;
      Bs[rr * 65 + cc] = (gk < K && gn < N) ? Bb[(long)gk * N + gn] : 0.f;
    }
    __syncthreads();
#pragma unroll
    for (int kk = 0; kk < 32; kk += 4) {
      int ko = kk + 2 * half;
      acc0 = wmma4(ldAf(As, t0m, lm, ko), ldBf(Bs, t0n, lm, ko), acc0);
      acc1 = wmma4(ldAf(As, t1m, lm, ko), ldBf(Bs, t1n, lm, ko), acc1);
    }
    __syncthreads();
  }
#pragma unroll
  for (int i = 0; i < 8; ++i) {
    int row = m0 + t0m * 16 + (half ? i + 8 : i), col = n0 + t0n * 16 + lm;
    if (row < M && col < N) Cb[(long)row * N + col] = acc0[i];
    row = m0 + t1m * 16 + (half ? i + 8 : i); col = n0 + t1n * 16 + lm;
    if (row < M && col < N) Cb[(long)row * N + col] = acc1[i];
  }
}

// batched complex GEMM, split re/im planes: Cre = ArBr - AiBi, Cim = ArBi + AiBr
__global__ __launch_bounds__(256) void k_gemm_cc(const float* __restrict__ A, const float* __restrict__ B,
                                                 float* __restrict__ C, int M, int N, int K,
                                                 long sA, long pA, long sB, long pB, long sC, long pC) {
  __shared__ float Ar[64 * 33], Ai[64 * 33], Br[32 * 65], Bi[32 * 65];
  const float* Arb = A + (long)blockIdx.z * sA;
  const float* Aib = Arb + pA;
  const float* Brb = B + (long)blockIdx.z * sB;
  const float* Bib = Brb + pB;
  float* Crb = C + (long)blockIdx.z * sC;
  float* Cib = Crb + pC;
  int m0 = blockIdx.y * 64, n0 = blockIdx.x * 64;
  int tid = threadIdx.x, lane = tid & 31, w = tid >> 5;
  int half = lane >> 4, lm = lane & 15;
  int tm[2] = {w >> 2, (w + 8) >> 2};
  int tn[2] = {w & 3, (w + 8) & 3};
  v8f acc[2][4];
#pragma unroll
  for (int t = 0; t < 2; ++t)
#pragma unroll
    for (int q = 0; q < 4; ++q) acc[t][q] = (v8f){0, 0, 0, 0, 0, 0, 0, 0};
  for (int kc = 0; kc < K; kc += 32) {
    for (int i = tid; i < 64 * 32; i += 256) {
      int rr = i >> 5, cc = i & 31;
      int gm = m0 + rr, gk = kc + cc;
      bool ok = (gm < M && gk < K);
      Ar[rr * 33 + cc] = ok ? Arb[(long)gm * K + gk] : 0.f;
      Ai[rr * 33 + cc] = ok ? Aib[(long)gm * K + gk] : 0.f;
    }
    for (int i = tid; i < 32 * 64; i += 256) {
      int rr = i >> 6, cc = i & 63;
      int gk = kc + rr, gn = n0 + cc;
      bool ok = (gk < K && gn < N);
      Br[rr * 65 + cc] = ok ? Brb[(long)gk * N + gn] : 0.f;
      Bi[rr * 65 + cc] = ok ? Bib[(long)gk * N + gn] : 0.f;
    }
    __syncthreads();
#pragma unroll
    for (int kk = 0; kk < 32; kk += 4) {
      int ko = kk + 2 * half;
#pragma unroll
      for (int t = 0; t < 2; ++t) {
        v2f ar = ldAf(Ar, tm[t], lm, ko), ai = ldAf(Ai, tm[t], lm, ko);
        v2f br = ldBf(Br, tn[t], lm, ko), bi = ldBf(Bi, tn[t], lm, ko);
        acc[t][0] = wmma4(ar, br, acc[t][0]);
        acc[t][1] = wmma4(ai, bi, acc[t][1]);
        acc[t][2] = wmma4(ar, bi, acc[t][2]);
        acc[t][3] = wmma4(ai, br, acc[t][3]);
      }
    }
    __syncthreads();
  }
#pragma unroll
  for (int t = 0; t < 2; ++t)
#pragma unroll
    for (int i = 0; i < 8; ++i) {
      int row = m0 + tm[t] * 16 + (half ? i + 8 : i), col = n0 + tn[t] * 16 + lm;
      if (row < M && col < N) {
        Crb[(long)row * N + col] = acc[t][0][i] - acc[t][1][i];
        Cib[(long)row * N + col] = acc[t][2][i] + acc[t][3][i];
      }
    }
}

// ---------------- layout packers ----------------
// xhat compact [s=(m,k)][l][(b,i)*2] -> planes [l][plane][b][m][i][k]
__global__ void k_pack_xhat(const float* __restrict__ Xc, float* __restrict__ P, int bo, int ci) {
  int Mc = 2 * bo - 1;
  long total = (long)Mc * Mc * bo * 32 * ci;
  long idx = (long)blockIdx.x * 256 + threadIdx.x;
  if (idx >= total) return;
  int i = idx % ci; long r = idx / ci;
  int b = r % 32; r /= 32;
  int l = r % bo; r /= bo;
  int k = r % Mc; r /= Mc;
  int m = (int)r;
  int Nc = 32 * ci * 2;
  long src = ((long)(m * Mc + k) * bo + l) * Nc + (b * ci + i) * 2;
  long Mrow = (long)32 * Mc, Kc = (long)ci * Mc;
  long d0 = ((long)(l * 2 + 0) * Mrow + (long)b * Mc + m) * Kc + (long)i * Mc + k;
  long d1 = ((long)(l * 2 + 1) * Mrow + (long)b * Mc + m) * Kc + (long)i * Mc + k;
  P[d0] = Xc[src]; P[d1] = Xc[src + 1];
}
// ohat planes [l][plane][r=(b,m)][c=(o,n)] -> [m][n][l][b][o] interleaved cplx
__global__ void k_pack_ohat(const float* __restrict__ Cc, float* __restrict__ P, int bo, int co) {
  int Mc = 2 * bo - 1;
  long M = 32L * Mc, N = (long)co * Mc;
  long total = (long)Mc * Mc * bo * 32 * co;
  long idx = (long)blockIdx.x * 256 + threadIdx.x;
  if (idx >= total) return;
  int o = idx % co; long r = idx / co;
  int b = r % 32; r /= 32;
  int l = r % bo; r /= bo;
  int n = r % Mc; r /= Mc;
  int m = (int)r;
  long rr = (long)b * Mc + m, cc = (long)o * Mc + n;
  P[idx * 2]     = Cc[((long)l * 2 + 0) * M * N + rr * N + cc];
  P[idx * 2 + 1] = Cc[((long)l * 2 + 1) * M * N + rr * N + cc];
}

// ---------------- tail: integrate + BN + MLP ----------------
__global__ void k_integrate(const float* __restrict__ h, float* __restrict__ out) {
  int idx = blockIdx.x * 256 + threadIdx.x;
  if (idx >= 32 * 64) return;
  int c = idx % 64, b = idx / 64;
  float s = 0.f;
  for (int j = 0; j < 6; ++j) {
    float wj = (float)dhw(j, 3);
    float t = 0.f;
    const float* hp = h + ((long)(b * 64 + c) * 6 + j) * 36;
    for (int ag = 0; ag < 36; ++ag) t += hp[ag];
    s += wj * t;
  }
  out[idx] = s * (1.0f / 72.0f); // (2pi/6)^2 / (8 pi^2)
}
__global__ void k_bn(const float* __restrict__ x, const float* __restrict__ g, const float* __restrict__ bb,
                     float* __restrict__ y, int R, int F) {
  int f = blockIdx.x * blockDim.x + threadIdx.x;
  if (f >= F) return;
  float mu = 0.f;
  for (int r = 0; r < R; ++r) mu += x[r * F + f];
  mu /= R;
  float var = 0.f;
  for (int r = 0; r < R; ++r) { float d = x[r * F + f] - mu; var += d * d; }
  var /= R;
  float inv = rsqrtf(var + 1e-5f);
  for (int r = 0; r < R; ++r) y[r * F + f] = g[f] * (x[r * F + f] - mu) * inv + bb[f];
}
__global__ void k_fc(const float* __restrict__ x, const float* __restrict__ w, const float* __restrict__ bias,
                     float* __restrict__ y, int R, int K, int N, int relu) {
  int idx = blockIdx.x * 256 + threadIdx.x;
  if (idx >= R * N) return;
  int n = idx % N, r = idx / N;
  float s = bias[n];
  for (int k = 0; k < K; ++k) s += x[r * K + k] * w[k * N + n];
  y[idx] = relu ? fmaxf(s, 0.f) : s;
}

#define LAUNCH(kern, n, ...)                                                             \
  do {                                                                                   \
    long _n = (long)(n);                                                                 \
    if (_n > 0) kern<<<dim3((unsigned)((_n + 255) / 256)), dim3(256), 0, stream>>>(__VA_ARGS__); \
  } while (0)

extern "C" void kernel_launch(void* const* d_in, const int* in_sizes, int n_in,
                              void* d_out, int out_size, void* d_ws, size_t ws_size,
                              hipStream_t stream) {
  (void)in_sizes; (void)n_in; (void)out_size; (void)ws_size;
  const float* x    = (const float*)d_in[0];
  const float* ks2  = (const float*)d_in[1];
  const float* kso3[6];
  for (int i = 0; i < 6; ++i) kso3[i] = (const float*)d_in[2 + i];
  const float* bn1g = (const float*)d_in[8];
  const float* bn1b = (const float*)d_in[9];
  const float* w1   = (const float*)d_in[10];
  const float* b1   = (const float*)d_in[11];
  const float* bn2g = (const float*)d_in[12];
  const float* bn2b = (const float*)d_in[13];
  const float* w2   = (const float*)d_in[14];
  const float* b2   = (const float*)d_in[15];
  const float* bn3g = (const float*)d_in[16];
  const float* bn3b = (const float*)d_in[17];
  const float* w3   = (const float*)d_in[18];
  const float* b3   = (const float*)d_in[19];

  char* base = (char*)d_ws;
  size_t off = 0;
  auto alloc = [&](size_t nbytes) -> void* {
    void* p = base + off;
    off = (off + nbytes + 255) & ~(size_t)255;
    return p;
  };

  // --- constants (recomputed each call; deterministic) ---
  double* lft = (double*)alloc(160 * sizeof(double));
  k_lftab<<<1, 160, 0, stream>>>(lft);

  float* WS2  = (float*)alloc(59L * 30 * 60 * 4);
  float* DLS2 = (float*)alloc(59L * 59 * 60 * 30 * 4);
  float* KHs2 = (float*)alloc(2L * 30 * 59 * 2 * 4);
  LAUNCH(k_ws2, 59L * 30 * 60, lft, WS2);
  LAUNCH(k_dls2, 59L * 59 * 60 * 30, lft, DLS2);
  LAUNCH(k_khs2, 2L * 30 * 59, ks2, lft, KHs2);

  const int BIs[6] = {30, 20, 14, 9, 6, 4};
  const int BOs[6] = {20, 14, 9, 6, 4, 3};
  const int CIs[6] = {2, 3, 6, 11, 20, 35};
  const int COs[6] = {3, 6, 11, 20, 35, 64};
  float *Wd[6], *DLc[6], *KH[6];
  for (int i = 0; i < 6; ++i) {
    int bi = BIs[i], bo = BOs[i], ci = CIs[i], co = COs[i], Mc = 2 * bo - 1;
    Wd[i]  = (float*)alloc((size_t)Mc * Mc * bo * 2 * bi * 4);
    DLc[i] = (float*)alloc((size_t)Mc * Mc * 2 * bo * bo * 4);
    KH[i]  = (float*)alloc((size_t)bo * 2 * ci * Mc * co * Mc * 4);
    LAUNCH(k_wd_so3, (long)Mc * Mc * bo * 2 * bi, lft, Wd[i], bi, bo);
    LAUNCH(k_dl_so3, (long)Mc * Mc * 2 * bo * bo, lft, DLc[i], bo);
    LAUNCH(k_khat_so3, (long)bo * ci * Mc * co * Mc, kso3[i], lft, KH[i], bo, ci, co);
  }

  // two ping-pong slots for the SO(3) signal h (max = 32*2*60^3 floats)
  size_t hbytes = (size_t)32 * 2 * 60 * 60 * 60 * 4;
  float* hA = (float*)alloc(hbytes);
  float* hB = (float*)alloc(hbytes);
  size_t scratch0 = off;

  // -------- S2 conv (b=30) --------
  {
    off = scratch0;
    float* xfc = (float*)alloc(59L * 60 * 32 * 2 * 4);
    float* xh  = (float*)alloc(59L * 30 * 64 * 4);
    float* fh  = (float*)alloc(59L * 59 * 30 * 128 * 4);
    float* G   = (float*)alloc(3481L * 60 * 128 * 4);
    float* T   = (float*)alloc(59L * 60 * 60 * 64 * 2 * 4);
    LAUNCH(k_dft_s2, 59L * 60 * 32, x, xfc);
    // xhat[m]: (30x64) = WS2[m](30x60) * xfc[m](60x64)
    k_gemm_rr<<<dim3(1, 1, 59), 256, 0, stream>>>(WS2, xfc, xh, 30, 64, 60, 1800L, 60L * 64, 30L * 64);
    LAUNCH(k_fhat_s2, 59L * 59 * 30 * 32 * 2, xh, KHs2, fh);
    // G[m,n]: (60x128) = DLS2[m,n](60x30) * fhat[m,n](30x128)
    k_gemm_rr<<<dim3(2, 1, 3481), 256, 0, stream>>>(DLS2, fh, G, 60, 128, 30, 1800L, 30L * 128, 60L * 128);
    LAUNCH(k_idft_n, 59L * 60 * 60 * 64, G, T, 60, 64);
    LAUNCH(k_idft_m_relu, 32L * 2 * 60 * 60 * 60, T, hA, 60, 2);
  }

  // -------- 6 SO(3) conv stages --------
  float* hin = hA;
  for (int i = 0; i < 6; ++i) {
    int bi = BIs[i], bo = BOs[i], ci = CIs[i], co = COs[i];
    int Ni = 2 * bi, No = 2 * bo, Mc = 2 * bo - 1;
    float* hout = (i % 2 == 0) ? hB : hA;
    off = scratch0;
    long Mrow = 32L * Mc, Ncol = (long)co * Mc, Kdim = (long)ci * Mc;
    float* U   = (float*)alloc((size_t)32 * ci * Ni * Ni * Mc * 2 * 4);
    float* xfc = (float*)alloc((size_t)Mc * Mc * Ni * 32 * ci * 2 * 4);
    float* xhc = (float*)alloc((size_t)Mc * Mc * bo * 32 * ci * 2 * 4);
    float* xhp = (float*)alloc((size_t)bo * 2 * Mrow * Kdim * 4);
    float* ohc = (float*)alloc((size_t)bo * 2 * Mrow * Ncol * 4);
    float* ohp = (float*)alloc((size_t)Mc * Mc * bo * 32 * co * 2 * 4);
    float* G   = (float*)alloc((size_t)Mc * Mc * No * 32 * co * 2 * 4);
    float* T   = (float*)alloc((size_t)Mc * No * No * 32 * co * 2 * 4);

    float scale = (float)((TWO_PI / Ni) * (TWO_PI / Ni));
    LAUNCH(k_dft_gamma, (long)32 * ci * Ni * Ni * Mc, hin, U, Ni, ci, Mc, scale);
    LAUNCH(k_dft_alpha2, (long)Mc * Mc * Ni * 32 * ci, U, xfc, Ni, ci, Mc);
    // xhat[m,k]: (bo x 64ci) = Wd[m,k](bo x 2bi) * xfc[m,k](2bi x 64ci)
    {
      int Ng = 64 * ci;
      k_gemm_rr<<<dim3((Ng + 63) / 64, 1, Mc * Mc), 256, 0, stream>>>(
          Wd[i], xfc, xhc, bo, Ng, Ni, (long)bo * Ni, (long)Ni * Ng, (long)bo * Ng);
    }
    LAUNCH(k_pack_xhat, (long)Mc * Mc * bo * 32 * ci, xhc, xhp, bo, ci);
    // ohat[l]: (32Mc x coMc) = xhat_l(32Mc x ciMc) * khat_l(ciMc x coMc), complex
    k_gemm_cc<<<dim3((unsigned)((Ncol + 63) / 64), (unsigned)((Mrow + 63) / 64), bo), 256, 0, stream>>>(
        xhp, KH[i], ohc, (int)Mrow, (int)Ncol, (int)Kdim,
        2L * Mrow * Kdim, (long)Mrow * Kdim,
        2L * Kdim * Ncol, (long)Kdim * Ncol,
        2L * Mrow * Ncol, (long)Mrow * Ncol);
    LAUNCH(k_pack_ohat, (long)Mc * Mc * bo * 32 * co, ohc, ohp, bo, co);
    // G[m,n]: (2bo x 64co) = DL[m,n](2bo x bo) * ohat[m,n](bo x 64co)
    {
      int Ng = 64 * co;
      k_gemm_rr<<<dim3((Ng + 63) / 64, 1, Mc * Mc), 256, 0, stream>>>(
          DLc[i], ohp, G, No, Ng, bo, (long)No * bo, (long)bo * Ng, (long)No * Ng);
    }
    LAUNCH(k_idft_n, (long)Mc * No * No * 32 * co, G, T, No, 32 * co);
    LAUNCH(k_idft_m_relu, (long)32 * co * No * No * No, T, hout, No, co);
    hin = hout;
  }

  // -------- integrate + BN/MLP head --------
  off = scratch0;
  float* feat = (float*)alloc(32 * 64 * 4);
  float* t1 = (float*)alloc(32 * 64 * 4);
  float* t2 = (float*)alloc(32 * 64 * 4);
  float* t3 = (float*)alloc(32 * 64 * 4);
  float* t4 = (float*)alloc(32 * 32 * 4);
  float* t5 = (float*)alloc(32 * 32 * 4);
  LAUNCH(k_integrate, 32L * 64, hin, feat);
  k_bn<<<1, 64, 0, stream>>>(feat, bn1g, bn1b, t1, 32, 64);
  LAUNCH(k_fc, 32L * 64, t1, w1, b1, t2, 32, 64, 64, 1);
  k_bn<<<1, 64, 0, stream>>>(t2, bn2g, bn2b, t3, 32, 64);
  LAUNCH(k_fc, 32L * 32, t3, w2, b2, t4, 32, 64, 32, 1);
  k_bn<<<1, 32, 0, stream>>>(t4, bn3g, bn3b, t5, 32, 32);
  LAUNCH(k_fc, 32L * 10, t5, w3, b3, (float*)d_out, 32, 32, 10, 0);
}